// EdgeDegreeEmbedding_10479720202729
// MI455X (gfx1250) — compile-verified
//
#include <hip/hip_runtime.h>

typedef __attribute__((ext_vector_type(2))) float v2f;
typedef __attribute__((ext_vector_type(8))) float v8f;

constexpr int kNCOEF = 25;   // (LMAX+1)^2
constexpr int kM0    = 5;    // LMAX+1 (m=0 coefficients)
constexpr int kC     = 128;  // channels
constexpr int kTILE  = 16;   // edges per block (= WMMA M)
constexpr float kCUTOFF  = 12.0f;
constexpr float kRESCALE = 5.0f;
constexpr float kEPS     = 1e-5f;

__device__ __forceinline__ float silu_f(float x) {
    return x / (1.0f + __expf(-x));
}

// LayerNorm + SiLU over 64 features, one thread per row (rows 0..15).
__device__ __forceinline__ void ln_silu_16x64(const float* __restrict__ hraw,
                                              float* __restrict__ hout,
                                              const float* __restrict__ g,
                                              const float* __restrict__ be,
                                              int tid) {
    if (tid < kTILE) {
        float s = 0.0f, s2 = 0.0f;
        #pragma unroll 8
        for (int c = 0; c < 64; ++c) {
            float v = hraw[tid * 64 + c];
            s += v; s2 += v * v;
        }
        float mu  = s * (1.0f / 64.0f);
        float var = s2 * (1.0f / 64.0f) - mu * mu;
        float inv = rsqrtf(var + kEPS);
        #pragma unroll 8
        for (int c = 0; c < 64; ++c) {
            float v = (hraw[tid * 64 + c] - mu) * inv * g[c] + be[c];
            hout[tid * 64 + c] = silu_f(v);
        }
    }
}

__global__ __launch_bounds__(128)
void edge_degree_embed_kernel(const float* __restrict__ x_edge,
                              const float* __restrict__ edge_distance,
                              const float* __restrict__ wigner_inv,
                              const int*   __restrict__ edge_index,
                              const float* __restrict__ w1, const float* __restrict__ b1,
                              const float* __restrict__ g1, const float* __restrict__ be1,
                              const float* __restrict__ w2, const float* __restrict__ b2,
                              const float* __restrict__ g2, const float* __restrict__ be2,
                              const float* __restrict__ w3, const float* __restrict__ b3,
                              const int*   __restrict__ node_offset_p,
                              float* __restrict__ out,
                              int E)
{
    // LDS: xe (8KB, reused for gathered wigner columns) + hraw 4KB + h1 4KB +
    //      h2 4KB + m0 40KB + envs  => ~60.3 KB
    __shared__ float xe[kTILE * 128];          // x_edge tile; later wig[16][25][5]
    __shared__ float hraw[kTILE * 64];         // pre-LN activations (both layers)
    __shared__ float h1[kTILE * 64];
    __shared__ float h2[kTILE * 64];
    __shared__ float m0s[kTILE * kM0 * kC];    // [16][640]
    __shared__ float envs[kTILE];

    const int tid   = threadIdx.x;
    const int wave  = tid >> 5;
    const int lane  = tid & 31;
    const int lrow  = lane & 15;          // M row (A/C) or N col (B/C) in tile
    const int khalf = (lane >> 4) << 1;   // 0 or 2: K sub-offset per A/B layout
    const int rbase = (lane >> 4) << 3;   // C/D: lanes>=16 hold rows M+8
    const int e0    = blockIdx.x * kTILE;

    // Early prefetch of the wigner rows this block will gather later.
    if (tid < kTILE && (e0 + tid) < E) {
        __builtin_prefetch(&wigner_inv[(size_t)(e0 + tid) * (kNCOEF * kNCOEF)], 0, 0);
    }

    // ---- Stage x_edge tile into LDS (zero-fill out-of-range rows) ----
    for (int i = tid; i < kTILE * 128; i += blockDim.x) {
        int e = e0 + (i >> 7);
        xe[i] = (e < E) ? x_edge[(size_t)e * 128 + (i & 127)] : 0.0f;
    }
    __syncthreads();

    // ---- Layer 1: [16,128] @ w1[128,64]  (wave w -> cols 16w..16w+15) ----
    {
        const int colBase = wave * 16;
        v8f acc = {};
        for (int k0 = 0; k0 < 128; k0 += 4) {
            v2f a, b;
            a.x = xe[lrow * 128 + k0 + khalf];
            a.y = xe[lrow * 128 + k0 + khalf + 1];
            b.x = w1[(k0 + khalf)     * 64 + colBase + lrow];
            b.y = w1[(k0 + khalf + 1) * 64 + colBase + lrow];
            acc = __builtin_amdgcn_wmma_f32_16x16x4_f32(
                false, a, false, b, (short)0, acc, false, false);
        }
        #pragma unroll
        for (int r = 0; r < 8; ++r) {
            int row = rbase + r;
            hraw[row * 64 + colBase + lrow] = acc[r] + b1[colBase + lrow];
        }
    }
    __syncthreads();
    ln_silu_16x64(hraw, h1, g1, be1, tid);
    __syncthreads();

    // ---- Layer 2: [16,64] @ w2[64,64] ----
    {
        const int colBase = wave * 16;
        v8f acc = {};
        for (int k0 = 0; k0 < 64; k0 += 4) {
            v2f a, b;
            a.x = h1[lrow * 64 + k0 + khalf];
            a.y = h1[lrow * 64 + k0 + khalf + 1];
            b.x = w2[(k0 + khalf)     * 64 + colBase + lrow];
            b.y = w2[(k0 + khalf + 1) * 64 + colBase + lrow];
            acc = __builtin_amdgcn_wmma_f32_16x16x4_f32(
                false, a, false, b, (short)0, acc, false, false);
        }
        #pragma unroll
        for (int r = 0; r < 8; ++r) {
            int row = rbase + r;
            hraw[row * 64 + colBase + lrow] = acc[r] + b2[colBase + lrow];
        }
    }
    __syncthreads();
    ln_silu_16x64(hraw, h2, g2, be2, tid);
    __syncthreads();

    // ---- Layer 3: [16,64] @ w3[64,640] -> m0 [16][5*128]  (40 col tiles / 4 waves) ----
    for (int t = wave; t < 40; t += 4) {
        const int colBase = t * 16;
        v8f acc = {};
        for (int k0 = 0; k0 < 64; k0 += 4) {
            v2f a, b;
            a.x = h2[lrow * 64 + k0 + khalf];
            a.y = h2[lrow * 64 + k0 + khalf + 1];
            b.x = w3[(size_t)(k0 + khalf)     * 640 + colBase + lrow];
            b.y = w3[(size_t)(k0 + khalf + 1) * 640 + colBase + lrow];
            acc = __builtin_amdgcn_wmma_f32_16x16x4_f32(
                false, a, false, b, (short)0, acc, false, false);
        }
        #pragma unroll
        for (int r = 0; r < 8; ++r) {
            int row = rbase + r;
            m0s[row * (kM0 * kC) + colBase + lrow] = acc[r] + b3[colBase + lrow];
        }
    }
    __syncthreads();

    // ---- Gather the 5 m=0 Wigner columns (col = l*(l+1)) into LDS (reuse xe) ----
    float* wig = xe;  // [16][25][5]
    for (int i = tid; i < kTILE * kNCOEF * kM0; i += blockDim.x) {
        int e_loc = i / (kNCOEF * kM0);
        int rem   = i % (kNCOEF * kM0);
        int r     = rem / kM0;
        int l     = rem % kM0;
        int e     = e0 + e_loc;
        wig[i] = (e < E)
            ? wigner_inv[(size_t)e * (kNCOEF * kNCOEF) + r * kNCOEF + l * (l + 1)]
            : 0.0f;
    }
    // Envelope (polynomial p=5) / RESCALE, per edge
    if (tid < kTILE) {
        int e = e0 + tid;
        float env = 0.0f;
        if (e < E) {
            float d = edge_distance[e] * (1.0f / kCUTOFF);
            if (d < 1.0f) {
                float d5 = d * d * d * d * d;
                env = 1.0f - 21.0f * d5 + 35.0f * d5 * d - 15.0f * d5 * d * d;
            }
        }
        envs[tid] = env * (1.0f / kRESCALE);
    }
    __syncthreads();

    // ---- Wigner contraction + scatter-add (c fastest -> coalesced atomics) ----
    const int noff = node_offset_p[0];
    for (int i = tid; i < kTILE * kNCOEF * kC; i += blockDim.x) {
        int e_loc = i / (kNCOEF * kC);
        int rem   = i % (kNCOEF * kC);
        int e     = e0 + e_loc;
        if (e >= E) continue;
        int r = rem >> 7;      // coefficient index 0..24
        int c = rem & 127;     // channel
        float s = 0.0f;
        #pragma unroll
        for (int l = 0; l < kM0; ++l) {
            s += wig[(e_loc * kNCOEF + r) * kM0 + l] * m0s[e_loc * (kM0 * kC) + l * kC + c];
        }
        int dst = edge_index[E + e] - noff;   // edge_index[1][e]
        atomicAdd(&out[((size_t)dst * kNCOEF + r) * kC + c], s * envs[e_loc]);
    }
}

extern "C" void kernel_launch(void* const* d_in, const int* in_sizes, int n_in,
                              void* d_out, int out_size, void* d_ws, size_t ws_size,
                              hipStream_t stream) {
    const float* x             = (const float*)d_in[0];
    const float* x_edge        = (const float*)d_in[1];
    const float* edge_distance = (const float*)d_in[2];
    const float* wigner_inv    = (const float*)d_in[3];
    const int*   edge_index    = (const int*)d_in[4];
    const float* w1  = (const float*)d_in[5];
    const float* b1  = (const float*)d_in[6];
    const float* g1  = (const float*)d_in[7];
    const float* be1 = (const float*)d_in[8];
    const float* w2  = (const float*)d_in[9];
    const float* b2  = (const float*)d_in[10];
    const float* g2  = (const float*)d_in[11];
    const float* be2 = (const float*)d_in[12];
    const float* w3  = (const float*)d_in[13];
    const float* b3  = (const float*)d_in[14];
    const int* node_offset = (const int*)d_in[15];

    const int E = in_sizes[1] / 128;

    // out = x (then edges atomically accumulate into it)
    hipMemcpyAsync(d_out, x, (size_t)out_size * sizeof(float),
                   hipMemcpyDeviceToDevice, stream);

    const int blocks = (E + kTILE - 1) / kTILE;
    edge_degree_embed_kernel<<<blocks, 128, 0, stream>>>(
        x_edge, edge_distance, wigner_inv, edge_index,
        w1, b1, g1, be1, w2, b2, g2, be2, w3, b3,
        node_offset, (float*)d_out, E);
}